// WindowedLSTMCell_10479720202312
// MI455X (gfx1250) — compile-verified
//
#include <hip/hip_runtime.h>
#include <stdint.h>

typedef __attribute__((ext_vector_type(2))) float v2f;
typedef __attribute__((ext_vector_type(8))) float v8f;

#define UNITS 1536
#define NGATE 6144      // 4*UNITS
#define CHAR_LEN 256
#define VOCAB 80
#define NMIX 10

__device__ __forceinline__ int imin(int a, int b) { return a < b ? a : b; }

// ---------------------------------------------------------------------------
// init: build x1 = [inputs(3), w_prev(80)]; seed inputs into x2/x3 heads
// ---------------------------------------------------------------------------
__global__ void wl_init(const float* __restrict__ inp, const float* __restrict__ w_prev,
                        float* __restrict__ x1, float* __restrict__ x2, float* __restrict__ x3) {
  int t = threadIdx.x;
  if (t < 3) { x1[t] = inp[t]; x2[t] = inp[t]; x3[t] = inp[t]; }
  if (t >= 3 && t < 83) x1[t] = w_prev[t - 3];
}

// ---------------------------------------------------------------------------
// Split-K GEMV over a concatenated vector [va(la) ; vb(lb)] against [Ma ; Mb],
// both (K x 6144) row-major. Block = 256 threads owning 1024 columns (float4),
// gridDim.x = 6 column tiles, gridDim.y = S k-splits.
// Vector tile is staged into LDS via CDNA5 async global->LDS loads.
// part[s][col] is written exactly once -> deterministic reduction later.
// ---------------------------------------------------------------------------
__global__ void wl_gemv_splitk(const float* __restrict__ va, int la, const float* __restrict__ Ma,
                               const float* __restrict__ vb, const float* __restrict__ Mb,
                               float* __restrict__ part, int K, int KT) {
  __shared__ float sv[256];
  const int t  = threadIdx.x;
  const int ct = blockIdx.x;   // column tile (0..5)
  const int s  = blockIdx.y;   // k split
  const int kbeg = s * KT;
  const int kend = imin(K, kbeg + KT);
  const int c4 = ct * 256 + t; // float4 column index (0..1535)

  float4 acc = make_float4(0.f, 0.f, 0.f, 0.f);

  for (int k0 = kbeg; k0 < kend; k0 += 256) {
    const int len = imin(256, kend - k0);
    __syncthreads();           // protect sv reuse across chunks
    const int k = k0 + t;
    if (t < len) {
      unsigned lds_off = (unsigned)(uintptr_t)(&sv[t]);  // low 32 bits = LDS addr
      if (k < la) {
        unsigned goff = (unsigned)k * 4u;
        asm volatile("global_load_async_to_lds_b32 %0, %1, %2"
                     :: "v"(lds_off), "v"(goff), "s"(va) : "memory");
      } else {
        unsigned goff = (unsigned)(k - la) * 4u;
        asm volatile("global_load_async_to_lds_b32 %0, %1, %2"
                     :: "v"(lds_off), "v"(goff), "s"(vb) : "memory");
      }
    }
    asm volatile("s_wait_asynccnt 0x0" ::: "memory");
    __syncthreads();

    for (int i = 0; i < len; ++i) {
      const int kk = k0 + i;
      const float* row = (kk < la) ? (Ma + (size_t)kk * NGATE)
                                   : (Mb + (size_t)(kk - la) * NGATE);
      const float4 w = reinterpret_cast<const float4*>(row)[c4];
      const float v = sv[i];
      acc.x += v * w.x; acc.y += v * w.y; acc.z += v * w.z; acc.w += v * w.w;
    }
  }
  reinterpret_cast<float4*>(part + (size_t)s * NGATE)[c4] = acc;
}

// ---------------------------------------------------------------------------
// Fused split-K reduction + LSTM gates. 1536 threads; gate order i,f,g,o.
// ---------------------------------------------------------------------------
__global__ void wl_gates(const float* __restrict__ part, int S,
                         const float* __restrict__ bias, const float* __restrict__ c_prev,
                         float* __restrict__ h_out, float* __restrict__ h_copy) {
  const int i = blockIdx.x * blockDim.x + threadIdx.x;  // 0..1535
  float zi = bias[i], zf = bias[i + UNITS], zg = bias[i + 2 * UNITS], zo = bias[i + 3 * UNITS];
  for (int s = 0; s < S; ++s) {  // fixed order -> deterministic
    const float* p = part + (size_t)s * NGATE;
    zi += p[i]; zf += p[i + UNITS]; zg += p[i + 2 * UNITS]; zo += p[i + 3 * UNITS];
  }
  const float gi = 1.f / (1.f + expf(-zi));
  const float gf = 1.f / (1.f + expf(-zf));
  const float gg = tanhf(zg);
  const float go = 1.f / (1.f + expf(-zo));
  const float c = gf * c_prev[i] + gi * gg;
  const float h = go * tanhf(c);
  h_out[i] = h;
  if (h_copy) h_copy[i] = h;
}

// ---------------------------------------------------------------------------
// Attention window: coef = h1n@Wd + bd; alpha/beta/kappa; phi; w = phi@sentence.
// w computed by wave 0 with v_wmma_f32_16x16x4_f32:
//   A (16x4) = phi[k0..k0+3] broadcast to every row -> every row of D equals
//   the partial GEMV for 16 columns; row M=0 (VGPR0, lanes 0..15) is the result.
// ---------------------------------------------------------------------------
__global__ void wl_window(const float* __restrict__ h1n,
                          const float* __restrict__ Wd, const float* __restrict__ bd,
                          const float* __restrict__ kappa_prev,
                          const float* __restrict__ sentence,
                          float* __restrict__ w_out1, float* __restrict__ w_out2) {
  __shared__ float red[256];
  __shared__ float coef[32];
  __shared__ float Am[NMIX], Bm[NMIX], Km[NMIX];
  __shared__ float PHI[CHAR_LEN];
  const int t = threadIdx.x;

  // coef[30] = h1n @ Wd + bd  (1536x30)
  float pm[30];
#pragma unroll
  for (int m = 0; m < 30; ++m) pm[m] = 0.f;
  for (int k = t; k < UNITS; k += 256) {
    const float hv = h1n[k];
    const float* wr = Wd + (size_t)k * 30;
#pragma unroll
    for (int m = 0; m < 30; ++m) pm[m] += hv * wr[m];
  }
  for (int m = 0; m < 30; ++m) {
    red[t] = pm[m];
    __syncthreads();
    for (int st = 128; st > 0; st >>= 1) {
      if (t < st) red[t] += red[t + st];
      __syncthreads();
    }
    if (t == 0) coef[m] = red[0] + bd[m];
    __syncthreads();
  }

  if (t < NMIX) {
    Am[t] = expf(coef[t]);
    Bm[t] = expf(coef[NMIX + t]);
    Km[t] = kappa_prev[t] + expf(coef[2 * NMIX + t]);
  }
  __syncthreads();

  {  // phi[u], u = 1..256
    const float u = (float)(t + 1);
    float p = 0.f;
#pragma unroll
    for (int m = 0; m < NMIX; ++m) {
      const float d = Km[m] - u;
      p += Am[m] * expf(-Bm[m] * d * d);
    }
    PHI[t] = p;
  }
  __syncthreads();

  // w = PHI @ sentence  via WMMA f32 16x16x4, wave 0 only (EXEC all ones)
  if (t < 32) {
    const int lane = t;
    const bool hi = lane >= 16;     // lanes 16..31 hold K+2/K+3 halves
    const int n = hi ? lane - 16 : lane;
    for (int j0 = 0; j0 < VOCAB; j0 += 16) {
      v8f acc = {};
      for (int k0 = 0; k0 < CHAR_LEN; k0 += 4) {
        v2f a, b;
        const int ka = hi ? k0 + 2 : k0;
        a[0] = PHI[ka];
        a[1] = PHI[ka + 1];
        b[0] = sentence[(size_t)ka * VOCAB + j0 + n];
        b[1] = sentence[(size_t)(ka + 1) * VOCAB + j0 + n];
        acc = __builtin_amdgcn_wmma_f32_16x16x4_f32(false, a, false, b,
                                                    (short)0, acc, false, false);
      }
      if (!hi) {  // row M=0 of D: VGPR0 across lanes 0..15
        const float val = acc[0];
        w_out1[j0 + n] = val;
        w_out2[j0 + n] = val;
      }
    }
  }
}

// ---------------------------------------------------------------------------
extern "C" void kernel_launch(void* const* d_in, const int* in_sizes, int n_in,
                              void* d_out, int out_size, void* d_ws, size_t ws_size,
                              hipStream_t stream) {
  const float* inp        = (const float*)d_in[0];
  const float* h1         = (const float*)d_in[1];
  const float* c1         = (const float*)d_in[2];
  const float* h2         = (const float*)d_in[3];
  const float* c2         = (const float*)d_in[4];
  const float* h3         = (const float*)d_in[5];
  const float* c3         = (const float*)d_in[6];
  const float* w_prev     = (const float*)d_in[7];
  const float* kappa_prev = (const float*)d_in[8];
  const float* sentence   = (const float*)d_in[9];
  const float* W1 = (const float*)d_in[10];
  const float* U1 = (const float*)d_in[11];
  const float* b1 = (const float*)d_in[12];
  const float* Wd = (const float*)d_in[13];
  const float* bd = (const float*)d_in[14];
  const float* W2 = (const float*)d_in[15];
  const float* U2 = (const float*)d_in[16];
  const float* b2 = (const float*)d_in[17];
  const float* W3 = (const float*)d_in[18];
  const float* U3 = (const float*)d_in[19];
  const float* b3 = (const float*)d_in[20];
  float* out = (float*)d_out;

  // ws layout (floats): x1@0(83), x2@96(1619), x3@1824(1619), partials@4096
  float* W  = (float*)d_ws;
  float* x1 = W + 0;
  float* x2 = W + 96;
  float* x3 = W + 1824;
  float* part = W + 4096;

  // split factor from available workspace (deterministic given sizes)
  const size_t availF = ws_size / 4;
  int S = 1;
  if (availF > (size_t)(4096 + NGATE)) {
    size_t s = (availF - 4096) / NGATE;
    S = (int)(s > 32 ? 32 : s);
    if (S < 1) S = 1;
  }
  const int K1 = 83 + UNITS;            // 1619
  const int K23 = 1619 + UNITS;         // 3155
  const int KT1 = (K1 + S - 1) / S;
  const int KT23 = (K23 + S - 1) / S;

  wl_init<<<1, 128, 0, stream>>>(inp, w_prev, x1, x2, x3);

  // LSTM 1
  wl_gemv_splitk<<<dim3(6, S), 256, 0, stream>>>(x1, 83, W1, h1, U1, part, K1, KT1);
  wl_gates<<<6, 256, 0, stream>>>(part, S, b1, c1, out + 0, x2 + 83);

  // attention window -> w into x2[3..82] and x3[3..82]
  wl_window<<<1, 256, 0, stream>>>(x2 + 83, Wd, bd, kappa_prev, sentence, x2 + 3, x3 + 3);

  // LSTM 2
  wl_gemv_splitk<<<dim3(6, S), 256, 0, stream>>>(x2, 1619, W2, h2, U2, part, K23, KT23);
  wl_gates<<<6, 256, 0, stream>>>(part, S, b2, c2, out + UNITS, x3 + 83);

  // LSTM 3
  wl_gemv_splitk<<<dim3(6, S), 256, 0, stream>>>(x3, 1619, W3, h3, U3, part, K23, KT23);
  wl_gates<<<6, 256, 0, stream>>>(part, S, b3, c3, out + 2 * UNITS, (float*)nullptr);
}